// HSE_53128745451774
// MI455X (gfx1250) — compile-verified
//
#include <hip/hip_runtime.h>
#include <hip/hip_bf16.h>

typedef __attribute__((ext_vector_type(2))) float v2f;
typedef __attribute__((ext_vector_type(8))) float v8f;

#define B_   32
#define H_   56
#define W_   56
#define C_   256
#define CR_  64
#define HW_  (H_ * W_)

// ---------------------------------------------------------------------------
// Stage 1: per-(image, spatial-chunk) partial sums for the global average
// pool. Deterministic (no float atomics) so graph replays are bit-identical.
// Threads 0..255 cover the 256 channels -> each iteration is one contiguous
// 1 KB row load per block.
// ---------------------------------------------------------------------------
__global__ void se_gap_partial(const float* __restrict__ x,
                               float* __restrict__ partial,
                               int chunks, int rowsPer) {
    const int b = blockIdx.y;       // image
    const int j = blockIdx.x;       // spatial chunk
    const int c = threadIdx.x;      // channel
    const int p0 = j * rowsPer;
    int p1 = p0 + rowsPer;
    if (p1 > HW_) p1 = HW_;
    const float* __restrict__ xp = x + ((size_t)b * HW_) * C_ + c;
    float sum = 0.0f;
    for (int p = p0; p < p1; ++p)
        sum += xp[(size_t)p * C_];
    partial[((size_t)(b * chunks + j)) * C_ + c] = sum;
}

// ---------------------------------------------------------------------------
// Stage 2: finish GAP reduction, then the SE MLP:
//   s[32,256] -> (s @ w1)[32,64] -> relu6 -> (@ w2)[32,256] -> hsigmoid -> g
// Single 256-thread block = 8 wave32 waves. Both GEMMs run on
// V_WMMA_F32_16X16X4_F32 (fp32 matrix pipe) so precision matches the fp32
// reference exactly; at ~2 MFLOP total the op count is irrelevant, this is
// about using the hardware matrix path correctly.
//
// VGPR layouts per CDNA5 ISA 7.12.2 (wave32):
//   A 16x4 f32 : lane m = lane%16, K = 2*(lane/16) + vgpr        (2 VGPRs)
//   B 4x16 f32 : N = lane%16,      K = 2*(lane/16) + vgpr        (2 VGPRs)
//   C/D 16x16  : N = lane%16,      M = vgpr + 8*(lane/16)        (8 VGPRs)
// ---------------------------------------------------------------------------
__global__ void se_mlp(const float* __restrict__ partial,
                       const float* __restrict__ w1,
                       const float* __restrict__ w2,
                       float* __restrict__ g,
                       int chunks) {
    __shared__ float s_lds[B_][C_];    // 32 KB
    __shared__ float h_lds[B_][CR_];   // 8 KB

    const int tid = threadIdx.x;

    // Finish the GAP reduction (deterministic fixed-order sum).
    for (int i = tid; i < B_ * C_; i += 256) {
        const int b = i >> 8;
        const int c = i & 255;
        float sum = 0.0f;
        for (int k = 0; k < chunks; ++k)
            sum += partial[((size_t)(b * chunks + k)) * C_ + c];
        s_lds[b][c] = sum * (1.0f / (float)HW_);
    }
    __syncthreads();

    const int wave = tid >> 5;
    const int lane = tid & 31;
    const int hh   = lane >> 4;   // 0/1: which half-wave
    const int ln   = lane & 15;

    // ---- GEMM1: s[32,256] x w1[256,64] -> relu6 -> h_lds[32,64] ----------
    {
        const int mt = wave >> 2;            // 0..1 : M tile
        const int nt = wave & 3;             // 0..3 : N tile
        v8f acc = {0.f, 0.f, 0.f, 0.f, 0.f, 0.f, 0.f, 0.f};
        for (int k0 = 0; k0 < C_; k0 += 4) {
            const int ka = k0 + 2 * hh;
            v2f a, bf;
            a.x  = s_lds[16 * mt + ln][ka];
            a.y  = s_lds[16 * mt + ln][ka + 1];
            bf.x = w1[(size_t)ka       * CR_ + 16 * nt + ln];
            bf.y = w1[(size_t)(ka + 1) * CR_ + 16 * nt + ln];
            acc = __builtin_amdgcn_wmma_f32_16x16x4_f32(
                false, a, false, bf, (short)0, acc, false, false);
        }
        #pragma unroll
        for (int v = 0; v < 8; ++v) {
            const float val = acc[v];
            h_lds[16 * mt + v + 8 * hh][16 * nt + ln] =
                fminf(fmaxf(val, 0.0f), 6.0f);   // ReLU6
        }
    }
    __syncthreads();

    // ---- GEMM2: h[32,64] x w2[64,256] -> hsigmoid -> g[32,256] -----------
    for (int i = 0; i < 4; ++i) {
        const int t  = wave * 4 + i;         // 0..31 output tiles
        const int mt = t >> 4;               // 0..1
        const int nt = t & 15;               // 0..15
        v8f acc = {0.f, 0.f, 0.f, 0.f, 0.f, 0.f, 0.f, 0.f};
        for (int k0 = 0; k0 < CR_; k0 += 4) {
            const int ka = k0 + 2 * hh;
            v2f a, bf;
            a.x  = h_lds[16 * mt + ln][ka];
            a.y  = h_lds[16 * mt + ln][ka + 1];
            bf.x = w2[(size_t)ka       * C_ + 16 * nt + ln];
            bf.y = w2[(size_t)(ka + 1) * C_ + 16 * nt + ln];
            acc = __builtin_amdgcn_wmma_f32_16x16x4_f32(
                false, a, false, bf, (short)0, acc, false, false);
        }
        #pragma unroll
        for (int v = 0; v < 8; ++v) {
            const float d = acc[v];
            const float gate = fminf(fmaxf(d + 3.0f, 0.0f), 6.0f) * (1.0f / 6.0f);
            g[(size_t)(16 * mt + v + 8 * hh) * C_ + 16 * nt + ln] = gate;
        }
    }
}

// ---------------------------------------------------------------------------
// Stage 3: out = x * g[b, c] broadcast. Pure streaming pass: 103 MB read +
// 103 MB write dominates the whole op. float4 (b128) loads/stores; the 32 KB
// gate table lives in L2/WGP$ so its re-reads are free.
// ---------------------------------------------------------------------------
__global__ void se_gate(const float4* __restrict__ x4,
                        const float4* __restrict__ g4,
                        float4* __restrict__ out4,
                        int total) {
    const int v = blockIdx.x * blockDim.x + threadIdx.x;
    if (v >= total) return;
    const int c4 = v & (C_ / 4 - 1);                 // 64 float4 per row
    const int b  = v / (HW_ * (C_ / 4));
    const float4 xx = x4[v];
    const float4 gg = g4[b * (C_ / 4) + c4];
    float4 oo;
    oo.x = xx.x * gg.x;
    oo.y = xx.y * gg.y;
    oo.z = xx.z * gg.z;
    oo.w = xx.w * gg.w;
    out4[v] = oo;
}

extern "C" void kernel_launch(void* const* d_in, const int* in_sizes, int n_in,
                              void* d_out, int out_size, void* d_ws, size_t ws_size,
                              hipStream_t stream) {
    const float* x  = (const float*)d_in[0];   // [32,56,56,256]
    const float* w1 = (const float*)d_in[1];   // [256,64]
    const float* w2 = (const float*)d_in[2];   // [64,256]
    float* out = (float*)d_out;

    // Workspace layout: g[32*256] first, then GAP partials.
    float* g = (float*)d_ws;
    const size_t g_bytes = (size_t)B_ * C_ * sizeof(float);
    float* partial = (float*)((char*)d_ws + g_bytes);

    // Pick as many spatial chunks (parallelism for the GAP pass) as the
    // workspace allows, capped at 49 (3136 = 49 * 64 rows per chunk).
    const size_t per_chunk = (size_t)B_ * C_ * sizeof(float);
    size_t avail = (ws_size > g_bytes) ? (ws_size - g_bytes) : 0;
    int chunks = (int)(avail / per_chunk);
    if (chunks < 1)  chunks = 1;
    if (chunks > 49) chunks = 49;
    const int rowsPer = (HW_ + chunks - 1) / chunks;

    // 1) GAP partial sums: (chunks x 32) blocks, coalesced channel reads.
    se_gap_partial<<<dim3(chunks, B_), 256, 0, stream>>>(x, partial, chunks, rowsPer);

    // 2) Reduce + SE MLP on fp32 WMMA, gate -> g.
    se_mlp<<<1, 256, 0, stream>>>(partial, w1, w2, g, chunks);

    // 3) Broadcast-multiply streaming pass.
    const int total4 = B_ * HW_ * (C_ / 4);   // 6,422,528 float4
    se_gate<<<(total4 + 255) / 256, 256, 0, stream>>>(
        (const float4*)x, (const float4*)g, (float4*)out, total4);
}